// SimTierLevel_34325378629727
// MI455X (gfx1250) — compile-verified
//
#include <hip/hip_runtime.h>
#include <math.h>

#ifndef __has_builtin
#define __has_builtin(x) 0
#endif

#define NCOLS      4096
#define K_BUCKETS  202
#define NDIM       4
#define ROWSTRIDE  (K_BUCKETS * NDIM)   // 808 floats = 3232 B (16B aligned)
#define BINS_PAD   208                  // 202 padded; 208 mod 64 = 16 -> copies rotate banks
#define NCOPIES    4
#define BLOCK      256

#if __has_builtin(__builtin_amdgcn_global_load_async_to_lds_b128)
#define HAVE_ASYNC_LDS 1
#else
#define HAVE_ASYNC_LDS 0
#endif

typedef int v4i __attribute__((ext_vector_type(4)));
typedef __attribute__((address_space(1))) v4i* g_v4i_p;   // prints as "int4 __device__*"
typedef __attribute__((address_space(3))) v4i* l_v4i_p;   // prints as "int4 __shared__*"

__global__ __launch_bounds__(BLOCK) void
SimTierLevel_hist_kernel(const float* __restrict__ cosine,
                         const float* __restrict__ emb,
                         float* __restrict__ out)
{
    __shared__ unsigned hist[NCOPIES * BINS_PAD];
    __shared__ float    semb[ROWSTRIDE];

    const int tid = threadIdx.x;
    const int row = blockIdx.x;

    // ---- Stage emb (808 floats = 202 x b128) into LDS.
    // CDNA5 path: async global->LDS copy tracked by ASYNCcnt (gfx1250-specific).
#if HAVE_ASYNC_LDS
    if (tid < K_BUCKETS) {
        __builtin_amdgcn_global_load_async_to_lds_b128(
            (g_v4i_p)(emb + tid * 4),
            (l_v4i_p)(&semb[tid * 4]),
            0, 0);
    }
#else
    if (tid < K_BUCKETS) {
        float4 v = *(const float4*)(emb + tid * 4);
        *(float4*)(&semb[tid * 4]) = v;
    }
#endif

    // ---- Zero the privatized histograms (4 copies x 208 bins).
    for (int i = tid; i < NCOPIES * BINS_PAD; i += BLOCK) hist[i] = 0u;
    __syncthreads();

    // Each thread uses one of 4 histogram copies to cut LDS atomic conflicts.
    unsigned* myhist = &hist[(tid & (NCOPIES - 1)) * BINS_PAD];

    // ---- Streaming phase: 4096 floats/row, 256 threads, 4 x float4 per thread.
    const float* rowp = cosine + (size_t)row * NCOLS;
#pragma unroll
    for (int i = 0; i < 4; ++i) {
        float4 v = *(const float4*)(rowp + (size_t)(i * BLOCK + tid) * 4);
        int i0 = (int)ceilf(v.x * 100.0f) + 100;
        int i1 = (int)ceilf(v.y * 100.0f) + 100;
        int i2 = (int)ceilf(v.z * 100.0f) + 100;
        int i3 = (int)ceilf(v.w * 100.0f) + 100;
        // Match JAX scatter-add semantics (OOB dropped) by clamping into range;
        // data in [0,1) stays in [100,200] anyway.
        i0 = min(max(i0, 0), K_BUCKETS - 1);
        i1 = min(max(i1, 0), K_BUCKETS - 1);
        i2 = min(max(i2, 0), K_BUCKETS - 1);
        i3 = min(max(i3, 0), K_BUCKETS - 1);
        atomicAdd(&myhist[i0], 1u);   // ds_add_u32 (no return)
        atomicAdd(&myhist[i1], 1u);
        atomicAdd(&myhist[i2], 1u);
        atomicAdd(&myhist[i3], 1u);
    }

#if HAVE_ASYNC_LDS
#if __has_builtin(__builtin_amdgcn_s_wait_asynccnt)
    __builtin_amdgcn_s_wait_asynccnt(0);
#else
    asm volatile("s_wait_asynccnt 0" ::: "memory");
#endif
#endif
    __syncthreads();

    // ---- Reduce copies + write output: out[row, k*4 + d] = log1p(count[k]) * emb[k*4+d].
    if (tid < K_BUCKETS) {
        unsigned c = hist[tid]
                   + hist[BINS_PAD + tid]
                   + hist[2 * BINS_PAD + tid]
                   + hist[3 * BINS_PAD + tid];
        float s = log1pf((float)c);
        float4 e = *(const float4*)(&semb[tid * 4]);
        float4 o;
        o.x = s * e.x;
        o.y = s * e.y;
        o.z = s * e.z;
        o.w = s * e.w;
        *(float4*)(out + (size_t)row * ROWSTRIDE + tid * 4) = o;  // aligned B128 store
    }
}

extern "C" void kernel_launch(void* const* d_in, const int* in_sizes, int n_in,
                              void* d_out, int out_size, void* d_ws, size_t ws_size,
                              hipStream_t stream) {
    (void)n_in; (void)out_size; (void)d_ws; (void)ws_size;
    const float* cosine = (const float*)d_in[0];
    const float* emb    = (const float*)d_in[1];
    float*       out    = (float*)d_out;

    const int B = in_sizes[0] / NCOLS;   // 8192 rows
    SimTierLevel_hist_kernel<<<dim3(B), dim3(BLOCK), 0, stream>>>(cosine, emb, out);
}